// CrystalGraphConvNet_38946763441060
// MI455X (gfx1250) — compile-verified
//
#include <hip/hip_runtime.h>
#include <math.h>

// ---------------- problem constants (from reference) ----------------
#define N_ATOM 60000
#define M_NBR  12
#define F0     92
#define FB     41
#define F_HID  64          // F
#define H_FC   128         // H
#define NCRYS  2000        // NC
#define NCONV  3
#define K_CONV 169         // 2F + FB
#define K_PAD  172         // padded to multiple of 4
#define KP2    (K_PAD / 2) // 86 row-pairs
#define C_CONV 128         // 2F
#define EPSBN  1e-5f

typedef __attribute__((ext_vector_type(2))) float v2f;
typedef __attribute__((ext_vector_type(8))) float v8f;

// ---------------- WMMA f32 16x16x4 wrapper ----------------
// A: 16x4, lanes 0-15 hold row(lane) K,K+1 ; lanes 16-31 hold row(lane&15) K+2,K+3
// B: 4x16, lanes 0-15 hold col(lane) K,K+1 ; lanes 16-31 hold col(lane&15) K+2,K+3
// C/D: VGPR i -> row i (lanes 0-15) / row 8+i (lanes 16-31), col = lane&15
__device__ __forceinline__ v8f wmma4(v2f a, v2f b, v8f c) {
    return __builtin_amdgcn_wmma_f32_16x16x4_f32(
        /*neg_a=*/false, a, /*neg_b=*/false, b,
        /*c_mod=*/(short)0, c, /*reuse_a=*/false, /*reuse_b=*/false);
}

__device__ __forceinline__ float sp_f(float x) {            // softplus
    return (x > 20.f) ? x : log1pf(expf(x));
}
__device__ __forceinline__ float sig_f(float x) {           // sigmoid
    return 1.f / (1.f + expf(-x));
}

// ---------------- embedding GEMM: (60000,92)@(92,64)+b ----------------
// LDS holds W as row-pairs: sWe[p*64 + c] = {W[2p][c], W[2p+1][c]}
__global__ __launch_bounds__(128) void embed_gemm(
    const float* __restrict__ A, const float* __restrict__ W,
    const float* __restrict__ bias, float* __restrict__ out) {
    __shared__ v2f sWe[(F0 / 2) * F_HID];                   // 46*64*8B = 23.5 KB
    int tid = threadIdx.x;
    for (int i = tid; i < (F0 / 2) * F_HID; i += 128) {
        int p = i >> 6, c = i & 63;                         // pair, column
        v2f w; w.x = W[(2 * p) * F_HID + c]; w.y = W[(2 * p + 1) * F_HID + c];
        sWe[i] = w;
    }
    __syncthreads();

    int wave = tid >> 5, lane = tid & 31;
    int rowTile = blockIdx.x * 4 + wave;
    if (rowTile >= (N_ATOM / 16)) return;
    int r0    = rowTile * 16;
    int col   = lane & 15;
    int khalf = (lane >> 4) << 1;                           // 0 or 2
    int phalf = lane >> 4;                                  // 0 or 1 (pair offset)
    const float* aRow = A + (size_t)(r0 + col) * F0;

    v8f acc[4];
    #pragma unroll
    for (int t = 0; t < 4; ++t) {
        float bv = bias[t * 16 + col];
        #pragma unroll
        for (int i = 0; i < 8; ++i) acc[t][i] = bv;
    }
    #pragma unroll 4
    for (int k = 0; k < F0; k += 4) {                       // 92 = 23 * 4
        int kk = k + khalf;
        int p  = (k >> 1) + phalf;
        v2f a; a.x = aRow[kk]; a.y = aRow[kk + 1];          // global_load_b64
        #pragma unroll
        for (int t = 0; t < 4; ++t) {
            v2f b = sWe[p * F_HID + t * 16 + col];          // single ds_load_b64
            acc[t] = wmma4(a, b, acc[t]);
        }
    }
    int rbase = r0 + ((lane >> 4) << 3);
    #pragma unroll
    for (int t = 0; t < 4; ++t)
        #pragma unroll
        for (int i = 0; i < 8; ++i)
            out[(size_t)(rbase + i) * F_HID + t * 16 + col] = acc[t][i];
}

// ---------------- fused gather+concat+GEMM: gated = total @ Wf + bf ----------------
__device__ __forceinline__ void mma8(const v2f* sW2, int p, int col,
                                     v2f a, v8f acc[8]) {
    #pragma unroll
    for (int t = 0; t < 8; ++t) {
        v2f b = sW2[p * C_CONV + t * 16 + col];             // single ds_load_b64
        acc[t] = wmma4(a, b, acc[t]);
    }
}

__global__ __launch_bounds__(128) void conv_gemm(
    const float* __restrict__ x, const float* __restrict__ nbrF,
    const int* __restrict__ nbrIdx, const float* __restrict__ Wf,
    const float* __restrict__ bf, float* __restrict__ gated) {
    __shared__ v2f sW2[KP2 * C_CONV];                       // 86*128*8B = 88 KB
    int tid = threadIdx.x;
    for (int i = tid; i < KP2 * C_CONV; i += 128) {
        int p = i >> 7, c = i & 127;                        // pair, column
        int r0 = 2 * p, r1 = 2 * p + 1;
        v2f w;
        w.x = (r0 < K_CONV) ? Wf[r0 * C_CONV + c] : 0.f;    // zero-pad K 169..171
        w.y = (r1 < K_CONV) ? Wf[r1 * C_CONV + c] : 0.f;
        sW2[i] = w;
    }
    __syncthreads();

    int wave = tid >> 5, lane = tid & 31;
    int rowTile = blockIdx.x * 4 + wave;
    if (rowTile >= (N_ATOM * M_NBR / 16)) return;
    int col   = lane & 15;
    int khalf = (lane >> 4) << 1;                           // 0 or 2
    int phalf = lane >> 4;                                  // 0 or 1

    // A-side row owned by this lane
    int r = rowTile * 16 + col;
    int n = r / M_NBR, m = r - n * M_NBR;
    const float* rs = x + (size_t)n * F_HID;                       // self features
    const float* rn = x + (size_t)nbrIdx[n * M_NBR + m] * F_HID;   // gathered nbr
    const float* rb = nbrF + (size_t)(n * M_NBR + m) * FB;         // bond features
    __builtin_prefetch(rb, 0, 1);                                  // global_prefetch_b8

    v8f acc[8];
    #pragma unroll
    for (int t = 0; t < 8; ++t) {
        float bv = bf[t * 16 + col];
        #pragma unroll
        for (int i = 0; i < 8; ++i) acc[t][i] = bv;
    }

    // phase 1: self-atom features, K = 0..63
    #pragma unroll 2
    for (int k = 0; k < 64; k += 4) {
        int kk = k + khalf;
        v2f a; a.x = rs[kk]; a.y = rs[kk + 1];
        mma8(sW2, (k >> 1) + phalf, col, a, acc);
    }
    // phase 2: neighbor-atom features, K = 64..127
    #pragma unroll 2
    for (int k = 64; k < 128; k += 4) {
        int kk = k + khalf;
        v2f a; a.x = rn[kk - 64]; a.y = rn[kk - 63];
        mma8(sW2, (k >> 1) + phalf, col, a, acc);
    }
    // phase 3: bond features, K = 128..167
    #pragma unroll 2
    for (int k = 128; k < 168; k += 4) {
        int kk = k + khalf;
        v2f a; a.x = rb[kk - 128]; a.y = rb[kk - 127];
        mma8(sW2, (k >> 1) + phalf, col, a, acc);
    }
    // tail: K = 168 valid, 169..171 are zero pad
    {
        v2f a; a.x = (phalf == 0) ? rb[40] : 0.f; a.y = 0.f;
        mma8(sW2, 84 + phalf, col, a, acc);
    }

    size_t rbase = (size_t)rowTile * 16 + ((lane >> 4) << 3);
    #pragma unroll
    for (int t = 0; t < 8; ++t)
        #pragma unroll
        for (int i = 0; i < 8; ++i)
            gated[(rbase + i) * C_CONV + t * 16 + col] = acc[t][i];
}

// ---------------- column sum / sum-of-squares (BN statistics) ----------------
__global__ __launch_bounds__(256) void col_stats(
    const float* __restrict__ X, int rows, int cols,
    float* __restrict__ gsum, float* __restrict__ gsq) {
    __shared__ float ssum[256], ssq[256];
    int tid = threadIdx.x;
    int c = tid % cols;                                     // cols divides 256
    int p = tid / cols;
    int P = 256 / cols;
    float s = 0.f, q = 0.f;
    for (long rr = (long)blockIdx.x * P + p; rr < rows; rr += (long)gridDim.x * P) {
        float v = X[rr * cols + c];
        s += v; q += v * v;
    }
    ssum[tid] = s; ssq[tid] = q;
    __syncthreads();
    if (tid < cols) {
        float ts = 0.f, tq = 0.f;
        for (int j = 0; j < P; ++j) { ts += ssum[j * cols + tid]; tq += ssq[j * cols + tid]; }
        atomicAdd(&gsum[tid], ts);
        atomicAdd(&gsq[tid],  tq);
    }
}

// ---------------- BN1 + sigmoid*softplus + sum over neighbors ----------------
__global__ __launch_bounds__(256) void bn_act_reduce(
    const float* __restrict__ gated, const float* __restrict__ sum1,
    const float* __restrict__ sq1, const float* __restrict__ g1,
    const float* __restrict__ be1, float* __restrict__ summed) {
    int gid = blockIdx.x * 256 + threadIdx.x;
    if (gid >= N_ATOM * F_HID) return;
    int n = gid / F_HID, f = gid - n * F_HID;
    const float inv = 1.f / (float)(N_ATOM * M_NBR);
    int fc = F_HID + f;
    float mF = sum1[f]  * inv, vF = sq1[f]  * inv - mF * mF;
    float mC = sum1[fc] * inv, vC = sq1[fc] * inv - mC * mC;
    float sF = rsqrtf(vF + EPSBN) * g1[f],  shF = be1[f]  - mF * sF;
    float sC = rsqrtf(vC + EPSBN) * g1[fc], shC = be1[fc] - mC * sC;
    const float* gp = gated + (size_t)n * M_NBR * C_CONV;
    float acc = 0.f;
    #pragma unroll
    for (int m = 0; m < M_NBR; ++m) {
        float vf = gp[m * C_CONV + f]         * sF + shF;
        float vc = gp[m * C_CONV + F_HID + f] * sC + shC;
        acc += sig_f(vf) * sp_f(vc);
    }
    summed[gid] = acc;
}

// ---------------- BN2 + residual + softplus ----------------
__global__ __launch_bounds__(256) void bn2_res(
    const float* __restrict__ x0, const float* __restrict__ summed,
    const float* __restrict__ sum2, const float* __restrict__ sq2,
    const float* __restrict__ g2, const float* __restrict__ be2,
    float* __restrict__ x1) {
    int gid = blockIdx.x * 256 + threadIdx.x;
    if (gid >= N_ATOM * F_HID) return;
    int f = gid % F_HID;
    const float inv = 1.f / (float)N_ATOM;
    float mu = sum2[f] * inv, va = sq2[f] * inv - mu * mu;
    float sc = rsqrtf(va + EPSBN) * g2[f], sh = be2[f] - mu * sc;
    x1[gid] = sp_f(x0[gid] + summed[gid] * sc + sh);
}

// ---------------- segment-sum pooling ----------------
__global__ __launch_bounds__(256) void pool_accum(
    const float* __restrict__ x, const int* __restrict__ cidx,
    float* __restrict__ csum, float* __restrict__ ccnt) {
    int gid = blockIdx.x * 256 + threadIdx.x;
    if (gid >= N_ATOM * F_HID) return;
    int n = gid / F_HID, f = gid - n * F_HID;
    int c = cidx[n];
    atomicAdd(&csum[(size_t)c * F_HID + f], x[gid]);
    if (f == 0) atomicAdd(&ccnt[c], 1.f);
}

// ---------------- head: mean -> softplus -> FC -> softplus -> out ----------------
__global__ __launch_bounds__(128) void head(
    const float* __restrict__ csum, const float* __restrict__ ccnt,
    const float* __restrict__ Wfc, const float* __restrict__ bfc,
    const float* __restrict__ Wout, const float* __restrict__ bout,
    float* __restrict__ out) {
    __shared__ float sc[F_HID];
    __shared__ float red[H_FC];
    int c = blockIdx.x, tid = threadIdx.x;
    if (tid < F_HID) {
        float cnt = fmaxf(ccnt[c], 1.f);
        sc[tid] = sp_f(csum[(size_t)c * F_HID + tid] / cnt);
    }
    __syncthreads();
    float h = bfc[tid];
    #pragma unroll 8
    for (int f = 0; f < F_HID; ++f) h += sc[f] * Wfc[f * H_FC + tid];
    h = sp_f(h);
    red[tid] = h * Wout[tid];
    __syncthreads();
    for (int s = H_FC / 2; s > 0; s >>= 1) {
        if (tid < s) red[tid] += red[tid + s];
        __syncthreads();
    }
    if (tid == 0) out[c] = red[0] + bout[0];
}

// ---------------- misc ----------------
__global__ __launch_bounds__(256) void zero_f32(float* __restrict__ p, long cnt) {
    long i = (long)blockIdx.x * 256 + threadIdx.x;
    if (i < cnt) p[i] = 0.f;
}

// ---------------- orchestration ----------------
extern "C" void kernel_launch(void* const* d_in, const int* in_sizes, int n_in,
                              void* d_out, int out_size, void* d_ws, size_t ws_size,
                              hipStream_t stream) {
    const float* atom_fea = (const float*)d_in[0];
    const float* nbr_fea  = (const float*)d_in[1];
    const int*   nbr_idx  = (const int*)  d_in[2];
    const int*   cidx     = (const int*)  d_in[3];
    const float* W_emb    = (const float*)d_in[4];
    const float* b_emb    = (const float*)d_in[5];
    const float* W_full   = (const float*)d_in[6];
    const float* b_full   = (const float*)d_in[7];
    const float* g1       = (const float*)d_in[8];
    const float* be1      = (const float*)d_in[9];
    const float* g2       = (const float*)d_in[10];
    const float* be2      = (const float*)d_in[11];
    const float* W_fc     = (const float*)d_in[12];
    const float* b_fc     = (const float*)d_in[13];
    const float* W_out    = (const float*)d_in[14];
    const float* b_out    = (const float*)d_in[15];
    float* out = (float*)d_out;

    // workspace carve-up (floats)
    float* ws = (float*)d_ws;
    size_t NF  = (size_t)N_ATOM * F_HID;                    // 3.84 M
    size_t GSZ = (size_t)N_ATOM * M_NBR * C_CONV;           // 92.16 M
    float* x0     = ws;
    float* x1     = ws + NF;
    float* summed = ws + 2 * NF;
    float* gated  = ws + 3 * NF;
    float* stat   = gated + GSZ;                            // 384 floats of BN stats
    float* sum1 = stat, *sq1 = stat + 128, *sum2 = stat + 256, *sq2 = stat + 320;
    float* csum = stat + 384;                               // NC*F
    float* ccnt = csum + (size_t)NCRYS * F_HID;             // NC

    // 1) embedding
    embed_gemm<<<(N_ATOM / 16 + 3) / 4, 128, 0, stream>>>(atom_fea, W_emb, b_emb, x0);

    // 2) conv layers
    float* xc = x0;
    float* xn = x1;
    for (int i = 0; i < NCONV; ++i) {
        zero_f32<<<2, 256, 0, stream>>>(stat, 384);
        conv_gemm<<<(N_ATOM * M_NBR / 16) / 4, 128, 0, stream>>>(
            xc, nbr_fea, nbr_idx,
            W_full + (size_t)i * K_CONV * C_CONV, b_full + i * C_CONV, gated);
        col_stats<<<1024, 256, 0, stream>>>(gated, N_ATOM * M_NBR, C_CONV, sum1, sq1);
        bn_act_reduce<<<(N_ATOM * F_HID + 255) / 256, 256, 0, stream>>>(
            gated, sum1, sq1, g1 + i * C_CONV, be1 + i * C_CONV, summed);
        col_stats<<<512, 256, 0, stream>>>(summed, N_ATOM, F_HID, sum2, sq2);
        bn2_res<<<(N_ATOM * F_HID + 255) / 256, 256, 0, stream>>>(
            xc, summed, sum2, sq2, g2 + i * F_HID, be2 + i * F_HID, xn);
        float* t = xc; xc = xn; xn = t;
    }

    // 3) pooling + head
    zero_f32<<<((NCRYS * F_HID + NCRYS) + 255) / 256, 256, 0, stream>>>(
        csum, (long)NCRYS * F_HID + NCRYS);
    pool_accum<<<(N_ATOM * F_HID + 255) / 256, 256, 0, stream>>>(xc, cidx, csum, ccnt);
    head<<<NCRYS, H_FC, 0, stream>>>(csum, ccnt, W_fc, b_fc, W_out, b_out, out);
}